// GATLayer_7516192768271
// MI455X (gfx1250) — compile-verified
//
#include <hip/hip_runtime.h>

typedef __attribute__((ext_vector_type(2))) float v2f;
typedef __attribute__((ext_vector_type(8))) float v8f;

#define NB   4096
#define NN   16
#define NA   8
#define DIN  128
#define DOBS 128
#define DOUTC 136  // DOBS + NA

// Kernel 0: v_src[i] = sum_o W_attn[o]    * W_fc[o,i]
//           v_dst[i] = sum_o W_attn[128+o]* W_fc[o,i]
__global__ __launch_bounds__(128) void prep_v_kernel(const float* __restrict__ Wfc,
                                                     const float* __restrict__ Wattn,
                                                     float* __restrict__ v_ws) {
  int i = threadIdx.x;  // 0..127
  float s1 = 0.f, s2 = 0.f;
  for (int o = 0; o < DIN; ++o) {
    float w = Wfc[o * DIN + i];   // coalesced across threads
    s1 = fmaf(Wattn[o], w, s1);
    s2 = fmaf(Wattn[DIN + o], w, s2);
  }
  v_ws[i] = s1;
  v_ws[DIN + i] = s2;
}

// Kernel 1: one block per batch b. Fuses attention logits (via f32 WMMA),
// sigmoid gating, masked-mean mix, and the big broadcast write.
__global__ __launch_bounds__(256) void gat_kernel(const float* __restrict__ h,
                                                  const float* __restrict__ pol,
                                                  const float* __restrict__ act,
                                                  const float* __restrict__ obs,
                                                  const float* __restrict__ v_ws,
                                                  float* __restrict__ out_obs,
                                                  float* __restrict__ out_w) {
  __shared__ __align__(16) float h_s[NN * DIN];     // 8 KB
  __shared__ __align__(16) float obs_s[NN * DOBS];  // 8 KB
  __shared__ __align__(16) float vsrc_s[DIN];
  __shared__ __align__(16) float vdst_s[DIN];
  __shared__ float pol_s[NN * NA];
  __shared__ float act_s[NN * NA];
  __shared__ float asrc_s[NN];
  __shared__ float adst_s[NN];
  __shared__ float w_s[NN * NN];
  __shared__ float sumz_s[NN * NA];
  __shared__ __align__(16) float outz_s[NN * NN * NA];  // 8 KB

  const int b = blockIdx.x;
  const int t = threadIdx.x;
  const float* hb = h   + (size_t)b * NN * DIN;
  const float* ob = obs + (size_t)b * NN * DOBS;
  const float* pb = pol + (size_t)b * NN * NA;
  const float* ab = act + (size_t)b * NN * NA;

  // ---- cooperative loads into LDS ----
  for (int idx = t; idx < NN * DIN; idx += 256)  h_s[idx]  = hb[idx];
  for (int idx = t; idx < NN * DOBS; idx += 256) obs_s[idx] = ob[idx];
  if (t < DIN)     { vsrc_s[t] = v_ws[t]; vdst_s[t] = v_ws[DIN + t]; }
  if (t < NN * NA) { pol_s[t] = pb[t];    act_s[t]  = ab[t]; }
  __syncthreads();

  // ---- wave 0: D(16x16) = A(16x128) x H^T(128x16) via V_WMMA_F32_16X16X4_F32.
  // Only A rows 0 (v_src) and 1 (v_dst) matter: D row m depends solely on A
  // row m, and we read only D rows 0/1. Lanes for rows >=2 load arbitrary
  // (vsrc) data with no masking — their D rows are never read. This keeps the
  // fragment build branch-free: one ds_load_b64 each for A and B per K-step.
  if (t < 32) {
    const int lane  = t;
    const int half2 = (lane >> 4) * 2;  // K-pair select within 4-wide K slab
    const int idx16 = lane & 15;        // A: row m ; B: col n
    const float* va     = (idx16 == 1) ? vdst_s : vsrc_s;  // uniform per lane
    const float* hrow   = &h_s[idx16 * DIN];               // B[k][n] = h[n][k]
    v8f c = {0.f, 0.f, 0.f, 0.f, 0.f, 0.f, 0.f, 0.f};
    #pragma unroll
    for (int k0 = 0; k0 < DIN; k0 += 4) {
      const int kk = k0 + half2;
      v2f a  = *(const v2f*)(va + kk);     // A[m][kk], A[m][kk+1]
      v2f bm = *(const v2f*)(hrow + kk);   // B[kk][n], B[kk+1][n]
      c = __builtin_amdgcn_wmma_f32_16x16x4_f32(false, a, false, bm,
                                                (short)0, c, false, false);
    }
    // D layout: VGPR m&7 holds row m, lanes 0-15 for m<8. Rows 0,1 -> c[0],c[1].
    if (lane < 16) {
      asrc_s[lane] = c[0];
      adst_s[lane] = c[1];
    }
  }
  __syncthreads();

  // ---- w[i][j] = sigmoid(leaky_relu(a_src[j] + a_dst[i], 0.01)) ----
  {
    const int i = t >> 4, j = t & 15;
    float e = asrc_s[j] + adst_s[i];
    e = (e > 0.f) ? e : 0.01f * e;
    const float w = 1.f / (1.f + __expf(-e));
    w_s[t] = w;
    out_w[(size_t)b * (NN * NN) + t] = w;  // second output, coalesced
  }
  __syncthreads();

  // ---- sum_z[i][a] = sum_j ( w[i][j]*act[j][a] + (1-w)*pol[j][a] ) ----
  if (t < NN * NA) {
    const int i = t >> 3, a = t & 7;
    float s = 0.f;
    #pragma unroll
    for (int j = 0; j < NN; ++j) {
      const float w = w_s[i * NN + j];
      s += w * act_s[j * NA + a] + (1.f - w) * pol_s[j * NA + a];
    }
    sumz_s[t] = s;
  }
  __syncthreads();

  // ---- out_z[i][k][a] = (sum_z[i][a] - zgate[i][k][a] + pol[k][a]) / N ----
  {
    const int i = t >> 4, k = t & 15;
    const float w = w_s[i * NN + k];
    #pragma unroll
    for (int a = 0; a < NA; ++a) {
      const float p  = pol_s[k * NA + a];
      const float zg = w * act_s[k * NA + a] + (1.f - w) * p;
      outz_s[t * NA + a] = (sumz_s[i * NA + a] - zg + p) * (1.f / NN);
    }
  }
  __syncthreads();

  // ---- big broadcast write: per (i,k) segment = 128 obs floats + 8 out_z
  //      floats = 34 float4; 256*34 float4 per block, fully coalesced ----
  const int NV4 = NN * NN * (DOUTC / 4);  // 8704
  float4* outp = (float4*)(out_obs + (size_t)b * NN * NN * DOUTC);
  for (int v = t; v < NV4; v += 256) {
    const int pair = v / 34;          // pair = i*16 + k
    const int c4   = v - pair * 34;
    const int k    = pair & 15;
    float4 val;
    if (c4 < 32) {
      val = *(const float4*)&obs_s[k * DOBS + c4 * 4];
    } else {
      const float* z = &outz_s[pair * NA + (c4 - 32) * 4];
      val = make_float4(z[0], z[1], z[2], z[3]);
    }
    outp[v] = val;   // global element offset: b*34816 + v*4
  }
}

extern "C" void kernel_launch(void* const* d_in, const int* in_sizes, int n_in,
                              void* d_out, int out_size, void* d_ws, size_t ws_size,
                              hipStream_t stream) {
  const float* h     = (const float*)d_in[0];
  const float* pol   = (const float*)d_in[1];
  const float* act   = (const float*)d_in[2];
  const float* obs   = (const float*)d_in[3];
  const float* Wfc   = (const float*)d_in[4];
  const float* Wattn = (const float*)d_in[5];

  float* out     = (float*)d_out;
  float* out_obs = out;                                          // [B*N, N, 136]
  float* out_w   = out + (size_t)NB * NN * NN * DOUTC;           // [B*N, N, 1]
  float* v_ws    = (float*)d_ws;                                 // 256 floats

  prep_v_kernel<<<1, 128, 0, stream>>>(Wfc, Wattn, v_ws);
  gat_kernel<<<NB, 256, 0, stream>>>(h, pol, act, obs, v_ws, out_obs, out_w);
}